// GCNLayer_1125281432194
// MI455X (gfx1250) — compile-verified
//
#include <hip/hip_runtime.h>

typedef __attribute__((ext_vector_type(2))) float v2f;
typedef __attribute__((ext_vector_type(8))) float v8f;

#define D 128  // D_IN == D_OUT == 128

// ---------------------------------------------------------------------------
// Degree / normalization kernels
// ---------------------------------------------------------------------------
__global__ void deg_init_kernel(float* __restrict__ deg, int n) {
    int i = blockIdx.x * blockDim.x + threadIdx.x;
    if (i < n) deg[i] = 1.0f;  // self-loop contributes 1 to every node
}

__global__ void deg_edges_kernel(const int* __restrict__ col, float* __restrict__ deg, int ne) {
    int e = blockIdx.x * blockDim.x + threadIdx.x;
    if (e < ne) unsafeAtomicAdd(&deg[col[e]], 1.0f);
}

__global__ void dinv_kernel(const float* __restrict__ deg, float* __restrict__ dinv, int n) {
    int i = blockIdx.x * blockDim.x + threadIdx.x;
    if (i < n) {
        float d = deg[i];
        dinv[i] = (d > 0.0f) ? __frsqrt_rn(d) : 0.0f;
    }
}

// ---------------------------------------------------------------------------
// GEMM: xw = x @ W  via V_WMMA_F32_16X16X4_F32 (wave32).
// One wave computes a 16-row x 128-col slab (8 accumulators of 16x16).
//
// W is staged in LDS in k-quad interleaved layout:
//   sW[(k>>2)*512 + n*4 + (k&3)] = W[k][n]
// so each lane's B pair (W[ka][n], W[ka+1][n]) is one contiguous 8-byte
// ds_load_b64, landing directly in an even-aligned VGPR pair for WMMA SRC1.
// Bank check: lanes 0-15 hit byte offsets ...+16m+0 (banks 4m,4m+1),
// lanes 16-31 hit ...+16m+8 (banks 4m+2,4m+3) -> conflict-free.
//
// Epilogue also writes the self-loop term dinv[r]^2 * xw[r] into `out`,
// fully initializing d_out for the subsequent atomic scatter.
// ---------------------------------------------------------------------------
__global__ void __launch_bounds__(256)
gemm_xw_kernel(const float* __restrict__ x, const float* __restrict__ W,
               const float* __restrict__ dinv, float* __restrict__ xw,
               float* __restrict__ out, int n) {
    __shared__ float sW[D * D];  // 64 KB of the 320 KB WGP LDS

    const int tid = threadIdx.x;
    // Stage W (coalesced global reads; LDS write conflicts are amortized away).
    for (int i = tid; i < D * D; i += 256) {
        const int k = i >> 7;
        const int nn = i & (D - 1);
        sW[(k >> 2) * 512 + nn * 4 + (k & 3)] = W[i];
    }
    __syncthreads();

    const int wave  = tid >> 5;
    const int lane  = tid & 31;
    const int mtile = blockIdx.x * 8 + wave;
    const int mbase = mtile * 16;
    if (mbase >= n) return;  // wave-uniform exit (EXEC stays all-ones for WMMA)

    const int m     = lane & 15;   // row within tile (A) / col within N-tile (B,C,D)
    const int khalf = lane >> 4;   // selects K pair {0,1} vs {2,3}

    v8f acc[8];
#pragma unroll
    for (int t = 0; t < 8; ++t) acc[t] = (v8f){0, 0, 0, 0, 0, 0, 0, 0};

    int xr = mbase + m;
    if (xr >= n) xr = n - 1;  // clamp (harmless dup load; stores are guarded)
    const float* xrow = x + (size_t)xr * D;

    for (int k0 = 0; k0 < D; k0 += 4) {
        // A 16x4: lane holds x[m][ka], x[m][ka+1], ka = k0 + 2*khalf
        v2f a = *(const v2f*)(xrow + k0 + 2 * khalf);
        // Base of this k-quad's interleaved B block; lane's pair at +8*khalf.
        const float* bq = sW + (k0 >> 2) * 512 + m * 4 + 2 * khalf;
#pragma unroll
        for (int t = 0; t < 8; ++t) {
            v2f b = *(const v2f*)(bq + t * 64);  // one ds_load_b64, aligned pair
            acc[t] = __builtin_amdgcn_wmma_f32_16x16x4_f32(
                false, a, false, b, (short)0, acc[t], false, false);
        }
    }

    // C/D layout: acc[t][i] -> row mbase + i + 8*khalf, col t*16 + m
#pragma unroll
    for (int i = 0; i < 8; ++i) {
        const int r = mbase + i + 8 * khalf;
        if (r < n) {
            const float s  = dinv[r];
            const float ss = s * s;
            float* xwr  = xw  + (size_t)r * D;
            float* outr = out + (size_t)r * D;
#pragma unroll
            for (int t = 0; t < 8; ++t) {
                const float v = acc[t][i];
                xwr[t * 16 + m]  = v;
                outr[t * 16 + m] = v * ss;  // self-loop term initializes agg
            }
        }
    }
}

// ---------------------------------------------------------------------------
// Edge scatter: one wave per edge; each lane moves 4 of 128 features.
// out[col] += dinv[row]*dinv[col] * xw[row]
// ---------------------------------------------------------------------------
__global__ void scatter_kernel(const int* __restrict__ row, const int* __restrict__ col,
                               const float* __restrict__ xw, const float* __restrict__ dinv,
                               float* __restrict__ out, int ne) {
    const size_t idx = (size_t)blockIdx.x * blockDim.x + threadIdx.x;
    const size_t e   = idx >> 5;  // 32 lanes share an edge
    if (e >= (size_t)ne) return;
    const int chunk = (int)(idx & 31);  // 4 floats per lane

    const int r = row[e];
    const int c = col[e];
    const float norm = dinv[r] * dinv[c];

    const float4 v = *(const float4*)(xw + (size_t)r * D + chunk * 4);
    float* o = out + (size_t)c * D + chunk * 4;
    unsafeAtomicAdd(o + 0, v.x * norm);
    unsafeAtomicAdd(o + 1, v.y * norm);
    unsafeAtomicAdd(o + 2, v.z * norm);
    unsafeAtomicAdd(o + 3, v.w * norm);
}

// ---------------------------------------------------------------------------
// out = relu(out + b), in place
// ---------------------------------------------------------------------------
__global__ void bias_relu_kernel(float* __restrict__ out, const float* __restrict__ b,
                                 size_t total) {
    const size_t i = (size_t)blockIdx.x * blockDim.x + threadIdx.x;
    if (i < total) {
        const float v = out[i] + b[i & (D - 1)];
        out[i] = v > 0.0f ? v : 0.0f;
    }
}

// ---------------------------------------------------------------------------
// Launcher
// ---------------------------------------------------------------------------
extern "C" void kernel_launch(void* const* d_in, const int* in_sizes, int n_in,
                              void* d_out, int out_size, void* d_ws, size_t ws_size,
                              hipStream_t stream) {
    const float* x  = (const float*)d_in[0];
    const int*   ei = (const int*)d_in[1];
    const float* W  = (const float*)d_in[2];
    const float* b  = (const float*)d_in[3];
    float* out = (float*)d_out;

    const int n  = in_sizes[0] / D;   // nodes
    const int ne = in_sizes[1] / 2;   // edges
    const int* row = ei;              // edge_index[0]
    const int* col = ei + ne;         // edge_index[1]

    // Workspace layout: xw [n*D] | deg [n] | dinv [n]
    float* xw   = (float*)d_ws;
    float* deg  = xw + (size_t)n * D;
    float* dinv = deg + n;

    const int B = 256;

    deg_init_kernel<<<(n + B - 1) / B, B, 0, stream>>>(deg, n);
    deg_edges_kernel<<<(ne + B - 1) / B, B, 0, stream>>>(col, deg, ne);
    dinv_kernel<<<(n + B - 1) / B, B, 0, stream>>>(deg, dinv, n);

    const int mtiles = (n + 15) / 16;
    gemm_xw_kernel<<<(mtiles + 7) / 8, B, 0, stream>>>(x, W, dinv, xw, out, n);

    const size_t scatter_threads = (size_t)ne * 32;
    scatter_kernel<<<(unsigned)((scatter_threads + B - 1) / B), B, 0, stream>>>(
        row, col, xw, dinv, out, ne);

    const size_t total = (size_t)n * D;
    bias_relu_kernel<<<(unsigned)((total + B - 1) / B), B, 0, stream>>>(out, b, total);
}